// Corr_60524679135699
// MI455X (gfx1250) — compile-verified
//
#include <hip/hip_runtime.h>
#include <cstdint>

// Problem constants (from reference)
#define MD     4
#define SS     9          // shifts per dim
#define NS     81         // total shifts
#define Bn     4
#define Cc     128
#define Hh     128
#define Ww     256
#define CSTRIDE (Hh * Ww) // channel stride in elements

#define ASTR   132        // padded LDS stride for A tile (conflict-free b64 reads)
#define NTHREAD 288       // 9 waves: one per dy

typedef __attribute__((ext_vector_type(2))) float v2f;
typedef __attribute__((ext_vector_type(8))) float v8f;

typedef __attribute__((address_space(1))) int gint_t;
typedef __attribute__((address_space(3))) int lint_t;

// ---- async global->LDS copy (CDNA5 ASYNCcnt path) ----
__device__ __forceinline__ void async_g2l_b32(const float* g, float* l) {
#if __has_builtin(__builtin_amdgcn_global_load_async_to_lds_b32)
  __builtin_amdgcn_global_load_async_to_lds_b32(
      (gint_t*)(uintptr_t)g,
      (lint_t*)(uintptr_t)l,
      0, 0);
#else
  asm volatile("global_load_async_to_lds_b32 %0, %1, off"
               :: "v"((unsigned)(uintptr_t)l), "v"(g)
               : "memory");
#endif
}

__device__ __forceinline__ void wait_async0() {
#if __has_builtin(__builtin_amdgcn_s_wait_asynccnt)
  __builtin_amdgcn_s_wait_asynccnt(0);
#else
  asm volatile("s_wait_asynccnt 0" ::: "memory");
#endif
}

// ---- K=128 accumulation: D[16x16] += A(16xK) * B(Kx16) via f32 WMMA ----
// bp   : points at this lane's B column (clamped to a legal address)
// arow : this lane's A row in LDS (As + n*ASTR)
// msk  : 0.0 for out-of-image columns (only used when MASKED)
template <bool MASKED>
__device__ __forceinline__ v8f band_mm(const float* __restrict__ bp,
                                       const float* __restrict__ arow,
                                       float msk, int mh) {
  v8f acc0 = {}, acc1 = {};
  for (int k0 = 0; k0 < Cc; k0 += 8) {
    const int kb = k0 + 2 * mh;
    // A frags: lanes 0-15 K=k0..k0+1 ; lanes 16-31 K=k0+2..k0+3
    v2f a0 = *(const v2f*)(arow + kb);
    v2f a1 = *(const v2f*)(arow + kb + 4);
    // B frags: this column of rows kb, kb+1 (and kb+4, kb+5)
    const float x0 = bp[(size_t)kb * CSTRIDE];
    const float x1 = bp[(size_t)(kb + 1) * CSTRIDE];
    const float x2 = bp[(size_t)(kb + 4) * CSTRIDE];
    const float x3 = bp[(size_t)(kb + 5) * CSTRIDE];
    v2f b0, b1;
    if (MASKED) {
      b0.x = x0 * msk; b0.y = x1 * msk;
      b1.x = x2 * msk; b1.y = x3 * msk;
    } else {
      b0.x = x0; b0.y = x1;
      b1.x = x2; b1.y = x3;
    }
    acc0 = __builtin_amdgcn_wmma_f32_16x16x4_f32(
        false, a0, false, b0, (short)0, acc0, false, false);
    acc1 = __builtin_amdgcn_wmma_f32_16x16x4_f32(
        false, a1, false, b1, (short)0, acc1, false, false);
  }
  return acc0 + acc1;
}

__global__ __launch_bounds__(NTHREAD)
void corr_wmma_kernel(const float* __restrict__ t1,
                      const float* __restrict__ t2,
                      float* __restrict__ out) {
  __shared__ float As[16 * ASTR];     // A tile: [m=16][c=128], padded
  __shared__ float outBuf[NS * 16];   // repack buffer: [s=81][m=16]

  const int tid = threadIdx.x;
  const int b  = blockIdx.z;
  const int h  = blockIdx.y;
  const int w0 = blockIdx.x * 16;

  // ---- stage A = t1[b, :, h, w0:w0+16] into LDS (transposed to [m][c]) ----
  {
    const float* gbase = t1 + (((size_t)b * Cc) * Hh + h) * Ww + w0;
    for (int e = tid; e < Cc * 16; e += NTHREAD) {
      const int m = e & 15;          // lanes contiguous in w -> coalesced lines
      const int c = e >> 4;
      async_g2l_b32(gbase + (size_t)c * CSTRIDE + m, &As[m * ASTR + c]);
    }
  }
  wait_async0();
  __syncthreads();

  const int lane = tid & 31;
  const int wv   = tid >> 5;   // 0..8 == dy
  const int n    = lane & 15;  // N column within tile / A fragment row m
  const int mh   = lane >> 4;  // half-wave: selects K pair (and output M half)

  // ---- banded GEMM: out[w, dx] = sum_c A[w,c] * t2row[c, w+dx-4] ----
  const int dy = wv;
  const int r  = h + dy - MD;
  const bool rvalid = (r >= 0) && (r < Hh);        // wave-uniform

#pragma unroll
  for (int t = 0; t < 2; ++t) {                    // two 16-wide N tiles
    const int u = w0 - MD + t * 16 + n;            // this lane's t2 column
    v8f acc = {};
    if (rvalid) {                                  // wave-uniform branch
      const bool uvalid = (u >= 0) && (u < Ww);    // per-lane
      // Column clipping only happens on the two border w-tiles -> uniform:
      const bool needMask = (t == 0) ? (w0 == 0) : (w0 == Ww - 16);
      const float* bp = t2 + (((size_t)b * Cc) * Hh + r) * Ww + (uvalid ? u : n);
      const float* arow = &As[n * ASTR];
      if (needMask)
        acc = band_mm<true>(bp, arow, uvalid ? 1.0f : 0.0f, mh);
      else
        acc = band_mm<false>(bp, arow, 1.0f, mh);
    }
    // extract the 9 diagonals: D[m][n] with dx = n - m (+16 for tile 1)
#pragma unroll
    for (int v = 0; v < 8; ++v) {
      const int m  = v + 8 * mh;                   // output row (w offset)
      const int dx = n - m + t * 16;
      if (dx >= 0 && dx < SS)
        outBuf[(dy * SS + dx) * 16 + m] = acc[v] * (1.0f / (float)Cc);
    }
  }
  __syncthreads();

  // ---- coalesced write: out[b, s, h, w0+m] ----
  for (int i = tid; i < NS * 16; i += NTHREAD) {
    const int s = i >> 4;
    const int m = i & 15;
    out[(((size_t)b * NS + s) * Hh + h) * Ww + w0 + m] = outBuf[i];
  }
}

extern "C" void kernel_launch(void* const* d_in, const int* in_sizes, int n_in,
                              void* d_out, int out_size, void* d_ws, size_t ws_size,
                              hipStream_t stream) {
  (void)in_sizes; (void)n_in; (void)out_size; (void)d_ws; (void)ws_size;
  const float* t1 = (const float*)d_in[0];
  const float* t2 = (const float*)d_in[1];
  float* out = (float*)d_out;

  dim3 grid(Ww / 16, Hh, Bn);   // (w-tile, h, b)
  dim3 block(NTHREAD);          // 9 waves of 32: one per dy
  corr_wmma_kernel<<<grid, block, 0, stream>>>(t1, t2, out);
}